// PSMLayer_36816459661730
// MI455X (gfx1250) — compile-verified
//
#include <hip/hip_runtime.h>

// Problem constants (match reference)
#define D_DIM   4096
#define N_DIM   1024
#define NNZ_PER 131072
#define K_FACT  4

typedef __attribute__((ext_vector_type(2))) float v2f;
typedef __attribute__((ext_vector_type(8))) float v8f;

// ---------------------------------------------------------------------------
// Zero-fill (float4 stores)
// ---------------------------------------------------------------------------
__global__ __launch_bounds__(256) void fill_zero_f4(float4* __restrict__ p, int n4) {
    int i = blockIdx.x * blockDim.x + threadIdx.x;
    if (i < n4) p[i] = make_float4(0.f, 0.f, 0.f, 0.f);
}

// ---------------------------------------------------------------------------
// 16x16 f32 tile transpose via WMMA identity trick:
//   dst[C,R] = src[R,C]^T  (+ optional bias indexed by the src-row dimension)
//
// Per wave: load tile T (rows r0..r0+15, cols c0..c0+15) in A-layout,
// accumulate D = sum_k A_k * E_k with V_WMMA_F32_16X16X4_F32 where E_k is the
// 4x16 identity slice selecting columns 4k..4k+3. D (C/D layout) holds T with
// lanes indexing columns -> transposed register layout -> contiguous stores.
// ---------------------------------------------------------------------------
__global__ __launch_bounds__(256)
void wmma_transpose(const float* __restrict__ src, float* __restrict__ dst,
                    const float* __restrict__ bias, int R, int C) {
    int waveId = (blockIdx.x * blockDim.x + threadIdx.x) >> 5;
    int lane   = threadIdx.x & 31;
    int tilesR = R >> 4;
    int r0 = (waveId % tilesR) << 4;
    int c0 = (waveId / tilesR) << 4;

    int  ln   = lane & 15;
    bool lo16 = lane < 16;
    int  hi   = lane >> 4;

    // Load one 16-float row of the tile per lane (lanes 0-15 and 16-31 read
    // the same rows; L0 broadcast absorbs the duplication).
    const float4* p = (const float4*)(src + (size_t)(r0 + ln) * C + c0);
    float t[16];
    float4 q;
    q = p[0]; t[0]  = q.x; t[1]  = q.y; t[2]  = q.z; t[3]  = q.w;
    q = p[1]; t[4]  = q.x; t[5]  = q.y; t[6]  = q.z; t[7]  = q.w;
    q = p[2]; t[8]  = q.x; t[9]  = q.y; t[10] = q.z; t[11] = q.w;
    q = p[3]; t[12] = q.x; t[13] = q.y; t[14] = q.z; t[15] = q.w;

    v8f c = {};
#pragma unroll
    for (int k = 0; k < 4; ++k) {
        v2f a, b;
        // A-layout (16x4): lanes 0-15 -> K=0 (v0), K=1 (v1); lanes 16-31 -> K=2,3
        a.x = lo16 ? t[4 * k + 0] : t[4 * k + 2];
        a.y = lo16 ? t[4 * k + 1] : t[4 * k + 3];
        // B-layout identity slice E_k[j,n] = (n == 4k+j)
        b.x = (ln == (lo16 ? 4 * k + 0 : 4 * k + 2)) ? 1.0f : 0.0f;
        b.y = (ln == (lo16 ? 4 * k + 1 : 4 * k + 3)) ? 1.0f : 0.0f;
        c = __builtin_amdgcn_wmma_f32_16x16x4_f32(false, a, false, b,
                                                  (short)0, c, false, false);
    }

    // c[p] = T[p + 8*hi, ln] -> dst[c0+ln, r0 + 8*hi + p], p = 0..7 contiguous
    float4 s0 = make_float4(c[0], c[1], c[2], c[3]);
    float4 s1 = make_float4(c[4], c[5], c[6], c[7]);
    if (bias) {
        const float4* bp = (const float4*)(bias + r0 + 8 * hi);
        float4 b0 = bp[0], b1 = bp[1];
        s0.x += b0.x; s0.y += b0.y; s0.z += b0.z; s0.w += b0.w;
        s1.x += b1.x; s1.y += b1.y; s1.z += b1.z; s1.w += b1.w;
    }
    float4* dp = (float4*)(dst + (size_t)(c0 + ln) * R + r0 + 8 * hi);
    dp[0] = s0;
    dp[1] = s1;
}

// ---------------------------------------------------------------------------
// COO scatter SpMM: Y[r,:] += v * X[c,:]
// One wave per nonzero batch (4 nnz per wave), 128-bit gathers over the 4KB
// source row, native f32 global atomics for the scatter, prefetch of the next
// nonzero's source row to hide random-gather latency.
// ---------------------------------------------------------------------------
__global__ __launch_bounds__(256)
void spmm_scatter(const float* __restrict__ vals, const int* __restrict__ rows,
                  const int* __restrict__ cols, const float* __restrict__ X,
                  float* __restrict__ Y) {
    int wave = (blockIdx.x * blockDim.x + threadIdx.x) >> 5;
    int lane = threadIdx.x & 31;
    int e0 = wave * 4;
#pragma unroll
    for (int t = 0; t < 4; ++t) {
        int   e = e0 + t;
        float v = vals[e];
        int   r = rows[e];
        int   ci = cols[e];
        if (t < 3) {
            // prefetch next source row: 32 lanes x 128B = full 4KB row
            const char* nxt = (const char*)(X + (size_t)cols[e + 1] * N_DIM);
            __builtin_prefetch(nxt + lane * 128, 0, 1);
        }
        const float4* srcRow = (const float4*)(X + (size_t)ci * N_DIM);
        float*        dstRow = Y + (size_t)r * N_DIM;
#pragma unroll
        for (int it = 0; it < N_DIM / 128; ++it) {  // 8 iterations
            int   f4 = it * 32 + lane;
            float4 x = srcRow[f4];
            int    b = f4 * 4;
            unsafeAtomicAdd(dstRow + b + 0, v * x.x);
            unsafeAtomicAdd(dstRow + b + 1, v * x.y);
            unsafeAtomicAdd(dstRow + b + 2, v * x.z);
            unsafeAtomicAdd(dstRow + b + 3, v * x.w);
        }
    }
}

// ---------------------------------------------------------------------------
// Launch: out = (S0 S1 S2 S3 U^T)^T + bias
// ---------------------------------------------------------------------------
extern "C" void kernel_launch(void* const* d_in, const int* in_sizes, int n_in,
                              void* d_out, int out_size, void* d_ws, size_t ws_size,
                              hipStream_t stream) {
    const float* U    = (const float*)d_in[0];  // [N, D]
    const float* bias = (const float*)d_in[1];  // [D]
    const float* vals = (const float*)d_in[2];  // [K, NNZ]
    const int*   rows = (const int*)d_in[3];    // [K, NNZ]
    const int*   cols = (const int*)d_in[4];    // [K, NNZ]
    float*       out  = (float*)d_out;          // [N, D]

    float* X0 = (float*)d_ws;                        // [D, N]
    float* X1 = X0 + (size_t)D_DIM * N_DIM;          // [D, N]

    dim3 blk(256);
    const int tiles      = (N_DIM / 16) * (D_DIM / 16);  // 16384 tiles
    const int tposBlocks = tiles / 8;                    // 8 waves per block
    const int n4         = D_DIM * N_DIM / 4;

    // X0 = U^T  (src [N, D] -> dst [D, N])
    wmma_transpose<<<tposBlocks, blk, 0, stream>>>(U, X0, nullptr, N_DIM, D_DIM);

    float* src = X0;
    float* dst = X1;
    for (int i = K_FACT - 1; i >= 0; --i) {
        fill_zero_f4<<<(n4 + 255) / 256, blk, 0, stream>>>((float4*)dst, n4);
        spmm_scatter<<<(NNZ_PER / 4) / 8, blk, 0, stream>>>(
            vals + (size_t)i * NNZ_PER, rows + (size_t)i * NNZ_PER,
            cols + (size_t)i * NNZ_PER, src, dst);
        float* tmp = src; src = dst; dst = tmp;
    }

    // out = y^T + bias  (src [D, N] -> dst [N, D])
    wmma_transpose<<<tposBlocks, blk, 0, stream>>>(src, out, bias, D_DIM, N_DIM);
}